// Space_EuclidDistance_Assign_Module_21612275434146
// MI455X (gfx1250) — compile-verified
//
#include <hip/hip_runtime.h>
#include <hip/hip_bf16.h>
#include <math.h>
#include <stdint.h>

typedef __attribute__((ext_vector_type(2))) float v2f;
typedef __attribute__((ext_vector_type(8))) float v8f;

// Problem constants
#define CC    256      // channels (batch dim of the GEMMs)
#define MM    128      // B*D
#define KK    784      // H*W
#define CN    128      // clusters
#define RTOT  100352   // B*D*H*W rows of x
#define KB    28       // K panel (784 = 28*28)
#define NPAN  (KK / KB)
#define PITCH 28       // contiguous; stride 28 mod 64 -> conflict-free 16-lane frag reads
#define PANEL (MM * PITCH)   // floats per panel buffer

// ---- CDNA5 async global->LDS copy (ASYNCcnt) ------------------------------
__device__ __forceinline__ void async_copy_b64(const float* gsrc, const float* ldst) {
    const unsigned ldsoff = (unsigned)(uintptr_t)ldst;   // addr[31:0] == LDS offset
    asm volatile("global_load_async_to_lds_b64 %0, %1, off"
                 :: "v"(ldsoff), "v"(gsrc) : "memory");
}
__device__ __forceinline__ void wait_async0() {
    asm volatile("s_wait_asynccnt 0x0" ::: "memory");
}

// ---------------------------------------------------------------------------
// Kernel 1: LayerNorm over C=256 + transpose [r=100352, c=256] -> xre[c][r]
// ---------------------------------------------------------------------------
__global__ __launch_bounds__(256) void ln_transpose(const float* __restrict__ x,
                                                    const float* __restrict__ lnw,
                                                    const float* __restrict__ lnb,
                                                    float* __restrict__ xre) {
    __shared__ __align__(16) float tile[256 * 33];  // [c][rr], pitch 33
    const int tid  = threadIdx.x;
    const int wave = tid >> 5;
    const int lane = tid & 31;
    const int r0   = blockIdx.x * 32;

    float lw[8], lb[8];
#pragma unroll
    for (int j = 0; j < 8; ++j) {
        lw[j] = lnw[lane + j * 32];
        lb[j] = lnb[lane + j * 32];
    }

#pragma unroll
    for (int p = 0; p < 4; ++p) {
        const int rr = p * 8 + wave;
        const int r  = r0 + rr;
        const float* row = x + (size_t)r * 256;
        float v[8], s = 0.f, sq = 0.f;
#pragma unroll
        for (int j = 0; j < 8; ++j) {
            v[j] = row[lane + j * 32];
            s  += v[j];
            sq += v[j] * v[j];
        }
#pragma unroll
        for (int off = 16; off > 0; off >>= 1) {
            s  += __shfl_xor(s, off, 32);
            sq += __shfl_xor(sq, off, 32);
        }
        const float mu   = s * (1.0f / 256.0f);
        const float var  = sq * (1.0f / 256.0f) - mu * mu;
        const float rstd = rsqrtf(var + 1e-5f);
#pragma unroll
        for (int j = 0; j < 8; ++j) {
            const int c = lane + j * 32;
            tile[c * 33 + rr] = (v[j] - mu) * rstd * lw[j] + lb[j];
        }
    }
    __syncthreads();

    const int rr = tid & 31;
    const int c0 = tid >> 5;
    for (int c = c0; c < 256; c += 8)
        xre[(size_t)c * RTOT + r0 + rr] = tile[c * 33 + rr];
}

// ---------------------------------------------------------------------------
// Kernel 2: row sum-of-squares over 784-float contiguous rows (one wave/row)
// ---------------------------------------------------------------------------
__global__ __launch_bounds__(256) void row_sumsq(const float* __restrict__ in,
                                                 float* __restrict__ out, int nrows) {
    const int row  = blockIdx.x * 8 + (threadIdx.x >> 5);
    const int lane = threadIdx.x & 31;
    if (row >= nrows) return;
    const float* p = in + (size_t)row * KK;
    float s = 0.f;
    for (int k = lane; k < KK; k += 32) { float v = p[k]; s += v * v; }
#pragma unroll
    for (int off = 16; off > 0; off >>= 1) s += __shfl_xor(s, off, 32);
    if (lane == 0) out[row] = s;
}

// ---------------------------------------------------------------------------
// Kernel 3: per-c GEMM x_re[c] (128x784) x cc[c]^T via f32 WMMA with
// double-buffered async-to-LDS panel staging; fused sqrt-dist + softmax.
// One block (8 waves) per c; wave w owns m-strip [16w,16w+16) x n=0..127.
// ---------------------------------------------------------------------------
__global__ __launch_bounds__(256) void gemm_xdist(const float* __restrict__ xre,
                                                  const float* __restrict__ cc,
                                                  const float* __restrict__ nx,
                                                  const float* __restrict__ ncc,
                                                  float* __restrict__ out_dist,
                                                  float* __restrict__ out_assign) {
    __shared__ __align__(16) float lA[2][PANEL];
    __shared__ __align__(16) float lB[2][PANEL];
    const int c    = blockIdx.x;
    const int tid  = threadIdx.x;
    const int wave = tid >> 5;
    const int lane = tid & 31;
    const int nl   = lane & 15;
    const int hsel = lane >> 4;
    const int ksel = hsel * 2;
    const int mbase = wave * 16;

    const float* Ap = xre + (size_t)c * (MM * KK);
    const float* Bp = cc  + (size_t)c * (CN * KK);

    // 128 rows x 14 b64-chunks = 1792 chunks = 256 threads x 7
    auto issue_panel = [&](int buf, int k0) {
#pragma unroll
        for (int i = 0; i < 7; ++i) {
            const int ch  = tid + i * 256;
            const int row = ch / 14;
            const int seg = (ch - row * 14) * 2;       // float offset within row
            async_copy_b64(Ap + (size_t)row * KK + k0 + seg, &lA[buf][row * PITCH + seg]);
            async_copy_b64(Bp + (size_t)row * KK + k0 + seg, &lB[buf][row * PITCH + seg]);
        }
    };

    v8f acc[8] = {};
    issue_panel(0, 0);
    wait_async0();
    __syncthreads();

    for (int p = 0; p < NPAN; ++p) {
        if (p + 1 < NPAN) issue_panel((p + 1) & 1, (p + 1) * KB);
        const float* la = &lA[p & 1][0];
        const float* lb = &lB[p & 1][0];
#pragma unroll
        for (int kk = 0; kk < KB; kk += 4) {
            v2f a = *(const v2f*)&la[(mbase + nl) * PITCH + kk + ksel];
#pragma unroll
            for (int j = 0; j < 8; ++j) {
                v2f b = *(const v2f*)&lb[(j * 16 + nl) * PITCH + kk + ksel];
                acc[j] = __builtin_amdgcn_wmma_f32_16x16x4_f32(
                    false, a, false, b, (short)0, acc[j], false, false);
            }
        }
        wait_async0();      // my next-panel copies landed (no-op on last iter)
        __syncthreads();    // everyone done reading buf[p&1] & next panel visible
    }

    float nxv[8];
#pragma unroll
    for (int v = 0; v < 8; ++v)
        nxv[v] = nx[c * MM + mbase + v + hsel * 8];

#pragma unroll
    for (int j = 0; j < 8; ++j) {
        const float nccv = ncc[c * CN + j * 16 + nl];
#pragma unroll
        for (int v = 0; v < 8; ++v) {
            const float d2 = nxv[v] + nccv - 2.0f * acc[j][v];
            const float d  = sqrtf(fmaxf(d2, 1e-12f));
            acc[j][v] = d;
            const int m = mbase + v + hsel * 8;
            const int n = j * 16 + nl;
            out_dist[(size_t)m * (CC * CN) + c * CN + n] = d;
        }
    }

    // fused softmax(-32*d) over n=0..127 (8 regs x 16-lane group)
#pragma unroll
    for (int v = 0; v < 8; ++v) {
        float dmin = acc[0][v];
#pragma unroll
        for (int j = 1; j < 8; ++j) dmin = fminf(dmin, acc[j][v]);
#pragma unroll
        for (int off = 1; off < 16; off <<= 1)
            dmin = fminf(dmin, __shfl_xor(dmin, off, 32));
        float e[8], s = 0.f;
#pragma unroll
        for (int j = 0; j < 8; ++j) {
            e[j] = __expf(-32.0f * (acc[j][v] - dmin));
            s += e[j];
        }
#pragma unroll
        for (int off = 1; off < 16; off <<= 1)
            s += __shfl_xor(s, off, 32);
        const float inv = 1.0f / s;
        const int m = mbase + v + hsel * 8;
#pragma unroll
        for (int j = 0; j < 8; ++j)
            out_assign[(size_t)m * (CC * CN) + c * CN + (j * 16 + nl)] = e[j] * inv;
    }
}

// ---------------------------------------------------------------------------
// Kernel 4: per-c self-cdist of cluster_center (A == B -> one panel stream)
// ---------------------------------------------------------------------------
__global__ __launch_bounds__(256) void gemm_cdist(const float* __restrict__ cc,
                                                  const float* __restrict__ ncc,
                                                  float* __restrict__ out) {
    __shared__ __align__(16) float lA[2][PANEL];
    const int c    = blockIdx.x;
    const int tid  = threadIdx.x;
    const int wave = tid >> 5;
    const int lane = tid & 31;
    const int nl   = lane & 15;
    const int hsel = lane >> 4;
    const int ksel = hsel * 2;
    const int mbase = wave * 16;

    const float* Ap = cc + (size_t)c * (CN * KK);

    auto issue_panel = [&](int buf, int k0) {
#pragma unroll
        for (int i = 0; i < 7; ++i) {
            const int ch  = tid + i * 256;
            const int row = ch / 14;
            const int seg = (ch - row * 14) * 2;
            async_copy_b64(Ap + (size_t)row * KK + k0 + seg, &lA[buf][row * PITCH + seg]);
        }
    };

    v8f acc[8] = {};
    issue_panel(0, 0);
    wait_async0();
    __syncthreads();

    for (int p = 0; p < NPAN; ++p) {
        if (p + 1 < NPAN) issue_panel((p + 1) & 1, (p + 1) * KB);
        const float* la = &lA[p & 1][0];
#pragma unroll
        for (int kk = 0; kk < KB; kk += 4) {
            v2f a = *(const v2f*)&la[(mbase + nl) * PITCH + kk + ksel];
#pragma unroll
            for (int j = 0; j < 8; ++j) {
                v2f b = *(const v2f*)&la[(j * 16 + nl) * PITCH + kk + ksel];
                acc[j] = __builtin_amdgcn_wmma_f32_16x16x4_f32(
                    false, a, false, b, (short)0, acc[j], false, false);
            }
        }
        wait_async0();
        __syncthreads();
    }

    float nmv[8];
#pragma unroll
    for (int v = 0; v < 8; ++v)
        nmv[v] = ncc[c * CN + mbase + v + hsel * 8];

#pragma unroll
    for (int j = 0; j < 8; ++j) {
        const float nnv = ncc[c * CN + j * 16 + nl];
#pragma unroll
        for (int v = 0; v < 8; ++v) {
            const float d2 = nmv[v] + nnv - 2.0f * acc[j][v];
            const float d  = sqrtf(fmaxf(d2, 1e-12f));
            const int m = mbase + v + hsel * 8;
            const int n = j * 16 + nl;
            out[(size_t)c * (CN * CN) + m * CN + n] = d;
        }
    }
}

// ---------------------------------------------------------------------------
extern "C" void kernel_launch(void* const* d_in, const int* in_sizes, int n_in,
                              void* d_out, int out_size, void* d_ws, size_t ws_size,
                              hipStream_t stream) {
    const float* x   = (const float*)d_in[0];   // [8,16,28,28,256]
    const float* lnw = (const float*)d_in[1];   // [256]
    const float* lnb = (const float*)d_in[2];   // [256]
    const float* cc  = (const float*)d_in[3];   // [256,128,784]

    float* out0 = (float*)d_out;                // x_distance        [128,256,128]
    float* out1 = out0 + (size_t)MM * CC * CN;  // x_distance_assign [128,256,128]
    float* out2 = out1 + (size_t)MM * CC * CN;  // cluster_dist      [256,128,128]

    float* xre = (float*)d_ws;                  // [256][100352] = 102.8 MB
    float* nx  = xre + (size_t)CC * RTOT;       // [256*128]
    float* ncc = nx + CC * MM;                  // [256*128]

    ln_transpose<<<RTOT / 32, 256, 0, stream>>>(x, lnw, lnb, xre);
    row_sumsq<<<(CC * MM) / 8, 256, 0, stream>>>(xre, nx, CC * MM);
    row_sumsq<<<(CC * CN) / 8, 256, 0, stream>>>(cc, ncc, CC * CN);
    gemm_xdist<<<CC, 256, 0, stream>>>(xre, cc, nx, ncc, out0, out1);
    gemm_cdist<<<CC, 256, 0, stream>>>(cc, ncc, out2);
}